// Decoder_80693845557763
// MI455X (gfx1250) — compile-verified
//
#include <hip/hip_runtime.h>
#include <hip/hip_bf16.h>

typedef __attribute__((ext_vector_type(2))) float v2f;
typedef __attribute__((ext_vector_type(8))) float v8f;

#define WMMA_F32(a, b, c) \
    __builtin_amdgcn_wmma_f32_16x16x4_f32(false, (a), false, (b), (short)0, (c), false, false)

__device__ __forceinline__ float lrelu(float x) {
    return x > 0.0f ? x : 0.01f * x;
}

// ---------------------------------------------------------------------------
// Generic GEMM: C[M,N] = act(A[M,K] (row stride lda) @ W[K,N] + bias[N])
// Wave tile: 16 (M) x 64 (N).  K stepped by 4 via v_wmma_f32_16x16x4_f32.
// Requires: M % 16 == 0, N % 64 == 0, K % 4 == 0.
// ---------------------------------------------------------------------------
template <int ACT>
__global__ __launch_bounds__(256) void gemm_bias_act(
    const float* __restrict__ A, int lda,
    const float* __restrict__ W,
    const float* __restrict__ bias,
    float* __restrict__ C,
    int M, int N, int K)
{
    const int wavesPerBlock = blockDim.x >> 5;
    const int wid = blockIdx.x * wavesPerBlock + (threadIdx.x >> 5);
    const int tilesN = N >> 6;
    const int tilesM = M >> 4;
    if (wid >= tilesM * tilesN) return;   // wave-uniform exit (EXEC stays full for WMMA)

    const int tm = wid / tilesN;
    const int tn = wid % tilesN;
    const int m0 = tm << 4;
    const int n0 = tn << 6;

    const int lane = threadIdx.x & 31;
    const int half = lane >> 4;     // 0: K+0/K+1, 1: K+2/K+3
    const int lr   = lane & 15;

    const float* __restrict__ Arow = A + (size_t)(m0 + lr) * lda;

    v8f acc0 = {}, acc1 = {}, acc2 = {}, acc3 = {};

    for (int k = 0; k < K; k += 4) {
        const int kk = k + 2 * half;
        v2f a;
        a.x = Arow[kk];
        a.y = Arow[kk + 1];

        const float* __restrict__ Wp0 = W + (size_t)kk * N + n0 + lr;
        const float* __restrict__ Wp1 = Wp0 + N;

        v2f b0; b0.x = Wp0[0];  b0.y = Wp1[0];
        v2f b1; b1.x = Wp0[16]; b1.y = Wp1[16];
        v2f b2; b2.x = Wp0[32]; b2.y = Wp1[32];
        v2f b3; b3.x = Wp0[48]; b3.y = Wp1[48];

        acc0 = WMMA_F32(a, b0, acc0);
        acc1 = WMMA_F32(a, b1, acc1);
        acc2 = WMMA_F32(a, b2, acc2);
        acc3 = WMMA_F32(a, b3, acc3);
    }

    // Store: VGPR v holds rows m0+v (lanes 0-15) / m0+v+8 (lanes 16-31)
    #pragma unroll
    for (int v = 0; v < 8; ++v) {
        const int row = m0 + v + 8 * half;
        float* __restrict__ Crow = C + (size_t)row * N;
        {
            const int c = n0 + lr;
            float x = acc0[v] + bias[c];
            Crow[c] = ACT ? lrelu(x) : x;
        }
        {
            const int c = n0 + 16 + lr;
            float x = acc1[v] + bias[c];
            Crow[c] = ACT ? lrelu(x) : x;
        }
        {
            const int c = n0 + 32 + lr;
            float x = acc2[v] + bias[c];
            Crow[c] = ACT ? lrelu(x) : x;
        }
        {
            const int c = n0 + 48 + lr;
            float x = acc3[v] + bias[c];
            Crow[c] = ACT ? lrelu(x) : x;
        }
    }
}

// ---------------------------------------------------------------------------
// Fused edge MLP: one workgroup = 16 upper-triangular pairs of one batch row.
//   feat[16,256] built in LDS, two 256->256 WMMA layers staged in LDS,
//   final 256->1 dot via LDS tree reduction, scattered to adj (both triangles).
// ---------------------------------------------------------------------------
#define EPAD 260                    // 256 + 4 floats padding (bank-conflict free)
#define EDGE_TILES 126              // 2016 pairs / 16

__global__ __launch_bounds__(256) void edge_mlp(
    const float* __restrict__ encNodes,   // [B,64,128]
    const float* __restrict__ encLat,     // [B,128]
    const float* __restrict__ w1, const float* __restrict__ b1,   // 256x256, 256
    const float* __restrict__ w2, const float* __restrict__ b2,   // 256x256, 256
    const float* __restrict__ w3, const float* __restrict__ b3,   // 256x1, 1
    float* __restrict__ adj)              // [B,64,64]
{
    __shared__ float sA[16 * EPAD];       // feat, later reused for z2
    __shared__ float sB[16 * EPAD];       // z1
    __shared__ float sred[16 * 16];
    __shared__ int sI[16], sJ[16];

    const int b    = blockIdx.x / EDGE_TILES;
    const int tile = blockIdx.x % EDGE_TILES;
    const int e0   = tile << 4;
    const int tid  = threadIdx.x;

    // pair indices for the 16 rows (jnp.triu_indices(64, k=1) row-major order)
    if (tid < 16) {
        int rem = e0 + tid, i = 0;
        while (rem >= 63 - i) { rem -= 63 - i; ++i; }
        sI[tid] = i;
        sJ[tid] = i + 1 + rem;
    }
    __syncthreads();

    // build feat tile: [:,0:128] = 0.5*(enc[i]+enc[j]), [:,128:256] = latent
    const float* __restrict__ EN = encNodes + (size_t)b * 64 * 128;
    const float* __restrict__ EL = encLat + (size_t)b * 128;
    #pragma unroll
    for (int it = 0; it < 16; ++it) {
        const int idx = tid + (it << 8);
        const int row = idx >> 8;
        const int col = idx & 255;
        float v;
        if (col < 128)
            v = 0.5f * (EN[sI[row] * 128 + col] + EN[sJ[row] * 128 + col]);
        else
            v = EL[col - 128];
        sA[row * EPAD + col] = v;
    }
    __syncthreads();

    const int lane = tid & 31;
    const int half = lane >> 4;
    const int lr   = lane & 15;
    const int w    = tid >> 5;      // wave id 0..7
    const int n0   = w << 5;        // 32 output columns per wave

    // ---- layer 1: sB = lrelu(sA @ w1 + b1) ----
    {
        v8f acc0 = {}, acc1 = {};
        for (int k = 0; k < 256; k += 4) {
            const int kk = k + 2 * half;
            v2f a;
            a.x = sA[lr * EPAD + kk];
            a.y = sA[lr * EPAD + kk + 1];
            const float* __restrict__ Wp0 = w1 + (size_t)kk * 256 + n0 + lr;
            const float* __restrict__ Wp1 = Wp0 + 256;
            v2f bb0; bb0.x = Wp0[0];  bb0.y = Wp1[0];
            v2f bb1; bb1.x = Wp0[16]; bb1.y = Wp1[16];
            acc0 = WMMA_F32(a, bb0, acc0);
            acc1 = WMMA_F32(a, bb1, acc1);
        }
        #pragma unroll
        for (int v = 0; v < 8; ++v) {
            const int row = v + 8 * half;
            const int c0 = n0 + lr, c1 = n0 + 16 + lr;
            sB[row * EPAD + c0] = lrelu(acc0[v] + b1[c0]);
            sB[row * EPAD + c1] = lrelu(acc1[v] + b1[c1]);
        }
    }
    __syncthreads();

    // ---- layer 2: sA = lrelu(sB @ w2 + b2)  (overwrite feat) ----
    {
        v8f acc0 = {}, acc1 = {};
        for (int k = 0; k < 256; k += 4) {
            const int kk = k + 2 * half;
            v2f a;
            a.x = sB[lr * EPAD + kk];
            a.y = sB[lr * EPAD + kk + 1];
            const float* __restrict__ Wp0 = w2 + (size_t)kk * 256 + n0 + lr;
            const float* __restrict__ Wp1 = Wp0 + 256;
            v2f bb0; bb0.x = Wp0[0];  bb0.y = Wp1[0];
            v2f bb1; bb1.x = Wp0[16]; bb1.y = Wp1[16];
            acc0 = WMMA_F32(a, bb0, acc0);
            acc1 = WMMA_F32(a, bb1, acc1);
        }
        #pragma unroll
        for (int v = 0; v < 8; ++v) {
            const int row = v + 8 * half;
            const int c0 = n0 + lr, c1 = n0 + 16 + lr;
            sA[row * EPAD + c0] = lrelu(acc0[v] + b2[c0]);
            sA[row * EPAD + c1] = lrelu(acc1[v] + b2[c1]);
        }
    }
    __syncthreads();

    // ---- final: vals[row] = z2[row,:] . w3 + b3 ----
    {
        const int row = tid >> 4;
        const int seg = tid & 15;
        float s = 0.0f;
        #pragma unroll
        for (int q = 0; q < 16; ++q)
            s += sA[row * EPAD + (seg << 4) + q] * w3[(seg << 4) + q];
        sred[row * 16 + seg] = s;
    }
    __syncthreads();

    if (tid < 16) {
        float v = b3[0];
        #pragma unroll
        for (int q = 0; q < 16; ++q) v += sred[tid * 16 + q];
        const int i = sI[tid], j = sJ[tid];
        float* __restrict__ Ab = adj + (size_t)b * 4096;
        Ab[i * 64 + j] = v;
        Ab[j * 64 + i] = v;
    }
}

__global__ void zero_diag(float* __restrict__ adj) {
    adj[(size_t)blockIdx.x * 4096 + threadIdx.x * 65] = 0.0f;
}

// ---------------------------------------------------------------------------

static inline int gemm_blocks(int M, int N) {
    const int waves = (M >> 4) * (N >> 6);
    return (waves + 7) / 8;   // 8 waves (256 threads) per block
}

extern "C" void kernel_launch(void* const* d_in, const int* in_sizes, int n_in,
                              void* d_out, int out_size, void* d_ws, size_t ws_size,
                              hipStream_t stream) {
    (void)in_sizes; (void)n_in; (void)out_size; (void)ws_size;

    const float* data  = (const float*)d_in[0];
    const float* nr_w1 = (const float*)d_in[1];  const float* nr_b1 = (const float*)d_in[2];
    const float* nr_w2 = (const float*)d_in[3];  const float* nr_b2 = (const float*)d_in[4];
    const float* nr_w3 = (const float*)d_in[5];  const float* nr_b3 = (const float*)d_in[6];
    const float* ne_w1 = (const float*)d_in[7];  const float* ne_b1 = (const float*)d_in[8];
    const float* ne_w2 = (const float*)d_in[9];  const float* ne_b2 = (const float*)d_in[10];
    const float* pl_w1 = (const float*)d_in[11]; const float* pl_b1 = (const float*)d_in[12];
    const float* pl_w2 = (const float*)d_in[13]; const float* pl_b2 = (const float*)d_in[14];
    const float* er_w1 = (const float*)d_in[15]; const float* er_b1 = (const float*)d_in[16];
    const float* er_w2 = (const float*)d_in[17]; const float* er_b2 = (const float*)d_in[18];
    const float* er_w3 = (const float*)d_in[19]; const float* er_b3 = (const float*)d_in[20];

    float* nodes = (float*)d_out;            // [128, 64*17] = 139264 floats
    float* adjM  = nodes + 128 * 1088;       // [128, 64, 64] = 524288 floats

    // workspace layout (floats)
    float* ws  = (float*)d_ws;
    float* H1  = ws;                 // 128 x  512
    float* H2  = H1  + 128 * 512;    // 128 x 1024
    float* P1  = H2  + 128 * 1024;   // 128 x  256
    float* EL  = P1  + 128 * 256;    // 128 x  128
    float* E1  = EL  + 128 * 128;    // 8192 x 128
    float* ENC = E1  + 8192 * 128;   // 8192 x 128   (total ~9.4 MB)

    const dim3 blk(256);

    // node reconstruction MLP
    gemm_bias_act<1><<<gemm_blocks(128, 512),  blk, 0, stream>>>(data, 128,  nr_w1, nr_b1, H1,    128, 512, 128);
    gemm_bias_act<1><<<gemm_blocks(128, 1024), blk, 0, stream>>>(H1,   512,  nr_w2, nr_b2, H2,    128, 1024, 512);
    gemm_bias_act<0><<<gemm_blocks(128, 1088), blk, 0, stream>>>(H2,   1024, nr_w3, nr_b3, nodes, 128, 1088, 1024);

    // edge prelayer
    gemm_bias_act<1><<<gemm_blocks(128, 256),  blk, 0, stream>>>(data, 128,  pl_w1, pl_b1, P1,    128, 256, 128);
    gemm_bias_act<1><<<gemm_blocks(128, 128),  blk, 0, stream>>>(P1,   256,  pl_w2, pl_b2, EL,    128, 128, 256);

    // per-node encoder: A = nodes with row stride 17 (drops last channel)
    gemm_bias_act<1><<<gemm_blocks(8192, 128), blk, 0, stream>>>(nodes, 17,  ne_w1, ne_b1, E1,   8192, 128, 16);
    gemm_bias_act<0><<<gemm_blocks(8192, 128), blk, 0, stream>>>(E1,   128,  ne_w2, ne_b2, ENC,  8192, 128, 128);

    // adjacency: zero diagonal, then fused edge MLP over all 2016 pairs x 128 batch
    zero_diag<<<128, 64, 0, stream>>>(adjM);
    edge_mlp<<<128 * EDGE_TILES, blk, 0, stream>>>(ENC, EL, er_w1, er_b1, er_w2, er_b2, er_w3, er_b3, adjM);
}